// LinkConv_44092134261313
// MI455X (gfx1250) — compile-verified
//
#include <hip/hip_runtime.h>
#include <hip/hip_bf16.h>

// CDNA5 / gfx1250 fused LinkConv.
// wave32; WMMA f32 16x16x4; fp32 global atomics; LDS weights in k-pair-swizzled
// layout so every WMMA B operand is one aligned ds_load_b64.

typedef __attribute__((ext_vector_type(2))) float v2f;
typedef __attribute__((ext_vector_type(8))) float v8f;

#define PSTRIDE 68   // 16x64 p-tile stored with padded row stride (bank-conflict avoidance)

__device__ __forceinline__ float atomic_add_f32(float* p, float v) {
    return __hip_atomic_fetch_add(p, v, __ATOMIC_RELAXED, __HIP_MEMORY_SCOPE_AGENT);
}

// ---------------------------------------------------------------------------
// Pass 1: input moments.  sum_x[16], sum_xxT[16][16] over nrows of a [nrows,16]
// fp32 matrix.  LDS-staged chunks of 256 rows; thread (k, rowgroup) accumulates
// the k-th slice of x·x^T over its 16 rows per chunk; block-reduce then atomics.
// ---------------------------------------------------------------------------
__global__ __launch_bounds__(256) void moments16_kernel(
    const float* __restrict__ x, int nrows,
    float* __restrict__ out_sx,   // [16]
    float* __restrict__ out_sxx)  // [16*16]
{
    __shared__ float sX[256 * 16];   // 16 KB
    const int tid = threadIdx.x;
    const int k  = tid & 15;
    const int rg = tid >> 4;         // 0..15

    float acc[16];
#pragma unroll
    for (int l = 0; l < 16; ++l) acc[l] = 0.f;
    float accx = 0.f;

    const long long total = (long long)nrows * 16;
    for (long long chunk = (long long)blockIdx.x * 256; chunk < nrows;
         chunk += (long long)gridDim.x * 256) {
        __syncthreads();
        for (int i = tid; i < 256 * 16; i += 256) {
            long long idx = chunk * 16 + i;
            sX[i] = (idx < total) ? x[idx] : 0.f;
        }
        __syncthreads();
#pragma unroll 1
        for (int rr = 0; rr < 16; ++rr) {
            const int r = rg * 16 + rr;
            const float xk = sX[r * 16 + k];
            accx += xk;
#pragma unroll
            for (int l = 0; l < 16; ++l) acc[l] += xk * sX[r * 16 + l];
        }
    }

    // block reduction (reuse sX)
    __syncthreads();
#pragma unroll
    for (int l = 0; l < 16; ++l) sX[tid * 16 + l] = acc[l];
    __syncthreads();
    {
        const int k2 = tid & 15, l2 = tid >> 4;
        float tot = 0.f;
#pragma unroll
        for (int g = 0; g < 16; ++g) tot += sX[(g * 16 + k2) * 16 + l2];
        atomic_add_f32(&out_sxx[k2 * 16 + l2], tot);
    }
    __syncthreads();
    sX[tid] = accx;
    __syncthreads();
    if (tid < 16) {
        float tot = 0.f;
#pragma unroll
        for (int g = 0; g < 16; ++g) tot += sX[g * 16 + tid];
        atomic_add_f32(&out_sx[tid], tot);
    }
}

// ---------------------------------------------------------------------------
// Pass 2: BN statistics of y = xW + b derived analytically from input moments:
//   mean_c = mu_x . W_c + b_c ;  var_c = W_c^T E[xx^T] W_c - (mu_x.W_c)^2
// ---------------------------------------------------------------------------
__global__ void prestats_kernel(const float* __restrict__ sx,
                                const float* __restrict__ sxx, float invN,
                                const float* __restrict__ Wpre,
                                const float* __restrict__ bpre,
                                float* __restrict__ out_mean,
                                float* __restrict__ out_inv)
{
    const int c = threadIdx.x;
    if (c >= 64) return;
    float wc[16];
    float m = 0.f;
#pragma unroll
    for (int k = 0; k < 16; ++k) {
        wc[k] = Wpre[k * 64 + c];
        m += (sx[k] * invN) * wc[k];
    }
    float ey2 = 0.f;
#pragma unroll
    for (int k = 0; k < 16; ++k) {
        float s = 0.f;
#pragma unroll
        for (int l = 0; l < 16; ++l) s += (sxx[k * 16 + l] * invN) * wc[l];
        ey2 += wc[k] * s;
    }
    out_mean[c] = m + bpre[c];
    out_inv[c]  = rsqrtf(fmaxf(ey2 - m * m, 0.f) + 1e-3f);
}

// ---------------------------------------------------------------------------
// Main fused kernel.  One wave owns a 16-row tile:
//   GEMM1: y = x @ Wpre  via 4(K) x 4(Ntile) V_WMMA_F32_16X16X4_F32
//   BN apply -> p staged in per-wave LDS (padded)
//   GEMM2: pw = p @ Wpos via 16(K) x 4(Ntile) WMMA
//   point_mode: scatter p*sin(pw), p*cos(pw) with fp32 global atomics
//   voxel_mode: gather accumulators, combine, write final, column sum/sumsq
//
// Weights staged in LDS in (k-pair, n, 2) order: element (k, n) lives at
//   sW[((k>>1)*64 + n)*2 + (k&1)]
// so the WMMA B operand (rows k0,k0+1 at column n) is one aligned ds_load_b64.
// ---------------------------------------------------------------------------
__global__ __launch_bounds__(256) void fused_branch_kernel(
    const float* __restrict__ X, int nrows,
    const int* __restrict__ unq_inv,
    const float* __restrict__ Wpre, const float* __restrict__ bpre,
    const float* __restrict__ gpre, const float* __restrict__ betap,
    const float* __restrict__ meanp, const float* __restrict__ invp,
    const float* __restrict__ Wpos, const float* __restrict__ bpos,
    float* __restrict__ acc_sin, float* __restrict__ acc_cos,
    float* __restrict__ out_final,
    float* __restrict__ col_sum, float* __restrict__ col_sumsq,
    int point_mode)
{
    __shared__ float sWpre[16 * 64];        //  4 KB, pair-swizzled
    __shared__ float sWpos[64 * 64];        // 16 KB, pair-swizzled
    __shared__ float sVec[3 * 64];          // scale_pre | shift_pre | b_pos
    __shared__ float sP[8][16 * PSTRIDE];   // ~34 KB, one 16x64 p tile per wave

    const int tid = threadIdx.x;
    // coalesced global read, pair-swizzled LDS scatter
    for (int i = tid; i < 16 * 64; i += 256) {
        const int kk = i >> 6, n = i & 63;
        sWpre[((kk >> 1) * 64 + n) * 2 + (kk & 1)] = Wpre[i];
    }
    for (int i = tid; i < 64 * 64; i += 256) {
        const int kk = i >> 6, n = i & 63;
        sWpos[((kk >> 1) * 64 + n) * 2 + (kk & 1)] = Wpos[i];
    }
    if (tid < 64) {
        const float sc = gpre[tid] * invp[tid];
        sVec[tid]       = sc;
        sVec[64 + tid]  = betap[tid] + sc * (bpre[tid] - meanp[tid]);
        sVec[128 + tid] = bpos[tid];
    }
    __syncthreads();

    const int wave = tid >> 5;       // wave32
    const int lane = tid & 31;
    const int lh   = lane >> 4;      // lane half (K split in A/B operand layout)
    const int l16  = lane & 15;
    float* __restrict__ myP = &sP[wave][0];

    float csum[4]   = {0.f, 0.f, 0.f, 0.f};
    float csumsq[4] = {0.f, 0.f, 0.f, 0.f};

    const long long ntiles = (long long)(nrows >> 4);
    const long long tstep  = (long long)gridDim.x * 8;
    for (long long t = (long long)blockIdx.x * 8 + wave; t < ntiles; t += tstep) {
        const long long base = t * 16;

        // prefetch next tile's A rows into cache (global_prefetch_b8)
        if (t + tstep < ntiles)
            __builtin_prefetch(X + ((size_t)((t + tstep) * 16 + l16)) * 16, 0, 0);

        // ---- GEMM1: 16x16x64 = x-tile @ Wpre ----
        v8f acc0[4] = {};
        const float* xrow = X + ((size_t)(base + l16)) * 16 + 2 * lh;
#pragma unroll
        for (int kk = 0; kk < 4; ++kk) {
            const v2f a = *(const v2f*)(xrow + 4 * kk);   // global_load_b64
            const int kh = 2 * kk + lh;                   // k-pair index
#pragma unroll
            for (int j = 0; j < 4; ++j) {
                const v2f b = *(const v2f*)&sWpre[(kh * 64 + 16 * j + l16) * 2];
                acc0[j] = __builtin_amdgcn_wmma_f32_16x16x4_f32(
                    false, a, false, b, (short)0, acc0[j], false, false);
            }
        }

        // ---- BN apply, stage p in LDS (row-major 16 x 64, stride 68) ----
#pragma unroll
        for (int j = 0; j < 4; ++j) {
            const int c = 16 * j + l16;
            const float sc = sVec[c], sh = sVec[64 + c];
#pragma unroll
            for (int r = 0; r < 8; ++r)
                myP[(r + 8 * lh) * PSTRIDE + c] = sc * acc0[j][r] + sh;
        }

        // ---- GEMM2: 16x64x64 = p-tile @ Wpos ----
        v8f acc2[4] = {};
#pragma unroll
        for (int kk = 0; kk < 16; ++kk) {
            const int kh = 2 * kk + lh;
            const v2f a = *(const v2f*)&myP[l16 * PSTRIDE + 4 * kk + 2 * lh];
#pragma unroll
            for (int j = 0; j < 4; ++j) {
                const v2f b = *(const v2f*)&sWpos[(kh * 64 + 16 * j + l16) * 2];
                acc2[j] = __builtin_amdgcn_wmma_f32_16x16x4_f32(
                    false, a, false, b, (short)0, acc2[j], false, false);
            }
        }

        if (point_mode) {
            int vox[8];
#pragma unroll
            for (int r = 0; r < 8; ++r) vox[r] = unq_inv[base + r + 8 * lh];
#pragma unroll
            for (int j = 0; j < 4; ++j) {
                const int c = 16 * j + l16;
                const float bp = sVec[128 + c];
#pragma unroll
                for (int r = 0; r < 8; ++r) {
                    const float pw = acc2[j][r] + bp;
                    const float pv = myP[(r + 8 * lh) * PSTRIDE + c];
                    float sv, cv;
                    __sincosf(pw, &sv, &cv);
                    atomic_add_f32(&acc_sin[(size_t)vox[r] * 64 + c], pv * sv);
                    atomic_add_f32(&acc_cos[(size_t)vox[r] * 64 + c], pv * cv);
                }
            }
        } else {
#pragma unroll
            for (int j = 0; j < 4; ++j) {
                const int c = 16 * j + l16;
                const float bp = sVec[128 + c];
#pragma unroll
                for (int r = 0; r < 8; ++r) {
                    const size_t row = (size_t)(base + r + 8 * lh);
                    const float mw = acc2[j][r] + bp;
                    const float fv = myP[(r + 8 * lh) * PSTRIDE + c];
                    float sv, cv;
                    __sincosf(mw, &sv, &cv);
                    const float cs = fv * sv, cc = fv * cv;
                    const float fin = (acc_sin[row * 64 + c] + cs) * cs +
                                      (acc_cos[row * 64 + c] + cc) * cc;
                    out_final[row * 64 + c] = fin;
                    csum[j]   += fin;
                    csumsq[j] += fin * fin;
                }
            }
        }
    }

    if (!point_mode) {
#pragma unroll
        for (int j = 0; j < 4; ++j) {
            const int c = 16 * j + l16;
            atomic_add_f32(&col_sum[c],   csum[j]);
            atomic_add_f32(&col_sumsq[c], csumsq[j]);
        }
    }
}

// Final BN stats over `final` columns.
__global__ void nstats_kernel(const float* __restrict__ sum,
                              const float* __restrict__ sumsq, float invV,
                              float* __restrict__ out_mean,
                              float* __restrict__ out_inv)
{
    const int c = threadIdx.x;
    if (c >= 64) return;
    const float m = sum[c] * invV;
    const float v = sumsq[c] * invV - m * m;
    out_mean[c] = m;
    out_inv[c]  = rsqrtf(fmaxf(v, 0.f) + 1e-3f);
}

// Elementwise BN + ReLU in place on d_out ([V,64]).
__global__ __launch_bounds__(256) void bnrelu_kernel(
    float* __restrict__ out, const float* __restrict__ mean,
    const float* __restrict__ inv, const float* __restrict__ gamma,
    const float* __restrict__ beta, long long total)
{
    for (long long i = (long long)blockIdx.x * blockDim.x + threadIdx.x;
         i < total; i += (long long)gridDim.x * blockDim.x) {
        const int c = (int)(i & 63);
        const float v = (out[i] - mean[c]) * inv[c] * gamma[c] + beta[c];
        out[i] = fmaxf(v, 0.f);
    }
}

extern "C" void kernel_launch(void* const* d_in, const int* in_sizes, int n_in,
                              void* d_out, int out_size, void* d_ws, size_t ws_size,
                              hipStream_t stream) {
    const float* feat = (const float*)d_in[0];   // [V,16]
    const float* pts  = (const float*)d_in[1];   // [N,16]
    const int*   unq  = (const int*)d_in[2];     // [N]
    const float* Wpre = (const float*)d_in[3];   // [16,64]
    const float* bpre = (const float*)d_in[4];   // [64]
    const float* gpre = (const float*)d_in[5];
    const float* betp = (const float*)d_in[6];
    const float* Wpos = (const float*)d_in[7];   // [64,64]
    const float* bpos = (const float*)d_in[8];
    const float* gn   = (const float*)d_in[9];
    const float* btn  = (const float*)d_in[10];

    const int V = in_sizes[0] / 16;
    const int N = in_sizes[1] / 16;

    // workspace layout
    float* ws       = (float*)d_ws;
    float* acc_sin  = ws;                               // V*64
    float* acc_cos  = acc_sin + (size_t)V * 64;         // V*64
    float* sx_p     = acc_cos + (size_t)V * 64;         // 16
    float* sxx_p    = sx_p + 16;                        // 256
    float* sx_f     = sxx_p + 256;                      // 16
    float* sxx_f    = sx_f + 16;                        // 256
    float* mean_p   = sxx_f + 256;                      // 64
    float* inv_p    = mean_p + 64;                      // 64
    float* mean_f   = inv_p + 64;                       // 64
    float* inv_f    = mean_f + 64;                      // 64
    float* csum     = inv_f + 64;                       // 64
    float* csumsq   = csum + 64;                        // 64
    float* mean_n   = csumsq + 64;                      // 64
    float* inv_n    = mean_n + 64;                      // 64

    const size_t zero_bytes =
        ((size_t)V * 64 * 2 + 16 + 256 + 16 + 256 + 64 * 8) * sizeof(float);
    hipMemsetAsync(d_ws, 0, zero_bytes, stream);

    moments16_kernel<<<256, 256, 0, stream>>>(pts, N, sx_p, sxx_p);
    moments16_kernel<<<128, 256, 0, stream>>>(feat, V, sx_f, sxx_f);
    prestats_kernel<<<1, 64, 0, stream>>>(sx_p, sxx_p, 1.0f / (float)N,
                                          Wpre, bpre, mean_p, inv_p);
    prestats_kernel<<<1, 64, 0, stream>>>(sx_f, sxx_f, 1.0f / (float)V,
                                          Wpre, bpre, mean_f, inv_f);

    // point branch: scatter-add into voxel accumulators
    fused_branch_kernel<<<1024, 256, 0, stream>>>(
        pts, N, unq, Wpre, bpre, gpre, betp, mean_p, inv_p, Wpos, bpos,
        acc_sin, acc_cos, nullptr, nullptr, nullptr, 1);

    // voxel branch: gather, combine, write `final` to d_out, column stats
    fused_branch_kernel<<<512, 256, 0, stream>>>(
        feat, V, nullptr, Wpre, bpre, gpre, betp, mean_f, inv_f, Wpos, bpos,
        acc_sin, acc_cos, (float*)d_out, csum, csumsq, 0);

    nstats_kernel<<<1, 64, 0, stream>>>(csum, csumsq, 1.0f / (float)V,
                                        mean_n, inv_n);
    bnrelu_kernel<<<2048, 256, 0, stream>>>((float*)d_out, mean_n, inv_n,
                                            gn, btn, (long long)V * 64);
}